// FilteredPatchLoss_87239375716985
// MI455X (gfx1250) — compile-verified
//
#include <hip/hip_runtime.h>

typedef __attribute__((ext_vector_type(2))) float v2f;
typedef __attribute__((ext_vector_type(4))) float v4f;
typedef __attribute__((ext_vector_type(8))) float v8f;

#define B_ 32
#define H_ 1024
#define W_ 1024
// PATCH = 16, nh = nw = 64, strips = B_*64*4 = 8192

__global__ void fpl_init(float* ws) {
    ws[0] = 0.0f;   // total valid loss
    ws[1] = 0.0f;   // valid count of last image
}

__global__ __launch_bounds__(256) void fpl_main(const float* __restrict__ outp,
                                                const float* __restrict__ tgtp,
                                                const float* __restrict__ frate,
                                                float* __restrict__ ws) {
    __shared__ float s_loss;
    __shared__ float s_cnt;
    if (threadIdx.x == 0) { s_loss = 0.0f; s_cnt = 0.0f; }
    __syncthreads();

    const int lane = threadIdx.x & 31;
    const int wave = threadIdx.x >> 5;
    const int s    = blockIdx.x * 8 + wave;   // strip id 0..8191
    const int pg   = s & 3;                   // group of 16 patches across width
    const int ph   = (s >> 2) & 63;           // patch row
    const int bb   = s >> 8;                  // image index

    const int m = lane & 15;                  // patch in strip == A-matrix row
    const int h = lane >> 4;                  // k-half (K=0,1 vs K=2,3)

    const int r0 = ph * 16 + h * 8;           // first of the 8 rows this lane reads
    const int c0 = pg * 256 + m * 16;         // column start of this lane's patch
    const size_t img = (size_t)bb * H_ * W_;

    v8f acc_l1 = {};                          // per-patch sum |o - t|
    v8f acc_t  = {};                          // per-patch sum t
    v2f ones;  ones.x = 1.0f; ones.y = 1.0f;  // B = all-ones 4x16 -> D rows = row sums of A

    #pragma unroll
    for (int rr = 0; rr < 8; ++rr) {
        const float* orow = outp + img + (size_t)(r0 + rr) * W_ + c0;
        const float* trow = tgtp + img + (size_t)(r0 + rr) * W_ + c0;
        if (rr < 6) {  // prefetch 2 rows ahead (global_prefetch_b8)
            __builtin_prefetch(orow + 2 * W_, 0, 3);
            __builtin_prefetch(trow + 2 * W_, 0, 3);
        }
        #pragma unroll
        for (int c = 0; c < 4; ++c) {
            v4f o = *(const v4f*)(orow + 4 * c);
            v4f t = *(const v4f*)(trow + 4 * c);
            v4f d;
            d.x = __builtin_fabsf(o.x - t.x);
            d.y = __builtin_fabsf(o.y - t.y);
            d.z = __builtin_fabsf(o.z - t.z);
            d.w = __builtin_fabsf(o.w - t.w);

            v2f a0; a0.x = d.x; a0.y = d.y;
            v2f a1; a1.x = d.z; a1.y = d.w;
            v2f b0; b0.x = t.x; b0.y = t.y;
            v2f b1; b1.x = t.z; b1.y = t.w;

            // D = A x ones + C : accumulates per-patch (row) sums in f32
            acc_l1 = __builtin_amdgcn_wmma_f32_16x16x4_f32(
                false, a0, false, ones, (short)0, acc_l1, false, false);
            acc_t  = __builtin_amdgcn_wmma_f32_16x16x4_f32(
                false, b0, false, ones, (short)0, acc_t,  false, false);
            acc_l1 = __builtin_amdgcn_wmma_f32_16x16x4_f32(
                false, a1, false, ones, (short)0, acc_l1, false, false);
            acc_t  = __builtin_amdgcn_wmma_f32_16x16x4_f32(
                false, b1, false, ones, (short)0, acc_t,  false, false);
        }
    }

    // C/D layout: lane l, VGPR v holds D[M = v + 8*h][N = l&15].
    // D[m][*] is the total sum for patch m (replicated over all 16 columns).
    const float thresh = frate[0] * 256.0f;   // mean > fr  <=>  sum > fr*256
    float ploss = 0.0f, pcnt = 0.0f;
    #pragma unroll
    for (int v = 0; v < 8; ++v) {
        float st = acc_t[v];
        float sl = acc_l1[v];
        if (st > thresh) {
            ploss += sl * (1.0f / 256.0f);    // per-patch L1 mean
            pcnt  += 1.0f;
        }
    }
    // take exactly one column: lane 0 (patches 0-7) and lane 16 (patches 8-15)
    if ((lane & 15) == 0) {
        atomicAdd(&s_loss, ploss);
        if (bb == (B_ - 1)) atomicAdd(&s_cnt, pcnt);
    }
    __syncthreads();
    if (threadIdx.x == 0) {
        atomicAdd(&ws[0], s_loss);
        atomicAdd(&ws[1], s_cnt);
    }
}

__global__ void fpl_fin(const float* ws, float* out) {
    out[0] = ws[0] / ws[1];
}

extern "C" void kernel_launch(void* const* d_in, const int* in_sizes, int n_in,
                              void* d_out, int out_size, void* d_ws, size_t ws_size,
                              hipStream_t stream) {
    const float* o  = (const float*)d_in[0];   // output [32,1024,1024] f32
    const float* t  = (const float*)d_in[1];   // target [32,1024,1024] f32
    // d_in[2]: patch_size (int scalar) -- layout hardcoded to 16
    const float* fr = (const float*)d_in[3];   // filter_rate (f32 scalar)
    float* ws  = (float*)d_ws;
    float* out = (float*)d_out;

    fpl_init<<<1, 1, 0, stream>>>(ws);
    fpl_main<<<1024, 256, 0, stream>>>(o, t, fr, ws);
    fpl_fin<<<1, 1, 0, stream>>>(ws, out);
}